// HierarchicalGraphSAGE_68023692034098
// MI455X (gfx1250) — compile-verified
//
#include <hip/hip_runtime.h>
#include <math.h>

typedef __attribute__((ext_vector_type(2))) float v2f;
typedef __attribute__((ext_vector_type(8))) float v8f;

// ---------------------------------------------------------------- utilities
__global__ void fill_f32(float* __restrict__ p, float val, size_t n) {
  size_t i = (size_t)blockIdx.x * blockDim.x + threadIdx.x;
  if (i < n) p[i] = val;
}

// ------------------------------------------------- edge scatter (mean aggr)
// one thread = one (edge, 4-feature chunk); float4 gather + 4 f32 atomics.
__global__ void scatter_mean_acc(const float* __restrict__ x,
                                 const int* __restrict__ src,
                                 const int* __restrict__ dst,
                                 const int* __restrict__ mask,   // null => all edges valid
                                 const int* __restrict__ remap,  // null => identity
                                 int E, int fshift,
                                 float* __restrict__ agg, float* __restrict__ cnt) {
  int tid = blockIdx.x * blockDim.x + threadIdx.x;
  int e = tid >> fshift;
  if (e >= E) return;
  int f4 = tid & ((1 << fshift) - 1);
  int s = src[e], d = dst[e];
  if (mask) {
    if (!(mask[s] && mask[d])) return;   // edge weight 0 -> no contribution
    s = remap[s];
    d = remap[d];
  }
  const int per = 1 << fshift;           // float4s per row
  const float4* xv = reinterpret_cast<const float4*>(x);
  float4 v = xv[(size_t)s * per + f4];
  float* ap = agg + ((size_t)d * per + f4) * 4;
  atomicAdd(ap + 0, v.x);
  atomicAdd(ap + 1, v.y);
  atomicAdd(ap + 2, v.z);
  atomicAdd(ap + 3, v.w);
  if (f4 == 0) atomicAdd(cnt + d, 1.0f);
}

// ---------------------------------------------- fused SAGE layer via WMMA
// y = l2norm((agg/max(cnt,1)) @ Wl + bl + x @ Wr); y = bn(y) + res; relu?
// res = x (din==128) or x @ projW + projb (refine layer 0, din==256).
// One wave owns a 16x128 output tile. K stepped by 4 with v_wmma_f32_16x16x4_f32.
// Loop nest: outer K, inner (unrolled) column tiles -> A loaded once per K-step
// as an aligned b64 load; invalid rows are zeroed by multiply (no EXEC games).
template <bool HASPROJ>
__global__ __launch_bounds__(256) void sage_layer_gemm(
    const float* __restrict__ agg, const float* __restrict__ cnt,
    const float* __restrict__ xroot,
    const float* __restrict__ Wl, const float* __restrict__ bl,
    const float* __restrict__ Wr,
    const float* __restrict__ gamma, const float* __restrict__ beta,
    const float* __restrict__ mean, const float* __restrict__ var,
    const float* __restrict__ xres,
    const float* __restrict__ projW, const float* __restrict__ projb,
    float* __restrict__ out, int n, int din, int do_relu) {
  const int wave = (blockIdx.x * blockDim.x + threadIdx.x) >> 5;
  const int lane = threadIdx.x & 31;
  const int m0 = wave * 16;
  if (m0 >= n) return;                 // wave-uniform branch (EXEC stays full)
  const int half = lane >> 4;          // 0: rows 0-7 / K lo, 1: rows 8-15 / K hi
  const int lm = lane & 15;
  const int koff = half * 2;           // A: lanes 16-31 carry K=k0+2,k0+3
  const int r = m0 + lm;               // row this lane feeds into A
  const bool rv = (r < n);
  const int rc = rv ? r : (n - 1);     // clamped: loads always in-bounds
  const float rvf = rv ? 1.f : 0.f;
  const float cv = cnt[rc];
  const float rcnt = rv ? (1.f / fmaxf(cv, 1.f)) : 0.f;  // also zeroes pad rows

  const float* __restrict__ agg_r = agg + (size_t)rc * din;
  const float* __restrict__ xrt_r = xroot + (size_t)rc * din;

  const v8f vzero = {0.f, 0.f, 0.f, 0.f, 0.f, 0.f, 0.f, 0.f};
  v8f acc[8];
  v8f accr[HASPROJ ? 8 : 1];
#pragma unroll
  for (int ct = 0; ct < 8; ++ct) acc[ct] = vzero;
  if (HASPROJ) {
#pragma unroll
    for (int ct = 0; ct < 8; ++ct) accr[ct] = vzero;
  }

  for (int k0 = 0; k0 < din; k0 += 4) {
    const int k = k0 + koff;                 // even -> 8B-aligned b64 loads
    v2f aA = *reinterpret_cast<const v2f*>(agg_r + k);
    aA *= rcnt;                              // mean + pad-row zeroing
    v2f aX = *reinterpret_cast<const v2f*>(xrt_r + k);
    aX *= rvf;                               // pad-row zeroing
    const float* __restrict__ wl = Wl + (size_t)k * 128 + lm;
    const float* __restrict__ wr = Wr + (size_t)k * 128 + lm;
#pragma unroll
    for (int ct = 0; ct < 8; ++ct) {
      v2f b0, b1;
      b0.x = wl[ct * 16];
      b0.y = wl[ct * 16 + 128];
      b1.x = wr[ct * 16];
      b1.y = wr[ct * 16 + 128];
      acc[ct] = __builtin_amdgcn_wmma_f32_16x16x4_f32(false, aA, false, b0,
                                                      (short)0, acc[ct], false,
                                                      false);
      acc[ct] = __builtin_amdgcn_wmma_f32_16x16x4_f32(false, aX, false, b1,
                                                      (short)0, acc[ct], false,
                                                      false);
    }
    if (HASPROJ) {
      const float* __restrict__ wp = projW + (size_t)k * 128 + lm;
#pragma unroll
      for (int ct = 0; ct < 8; ++ct) {
        v2f b2;
        b2.x = wp[ct * 16];
        b2.y = wp[ct * 16 + 128];
        accr[ct] = __builtin_amdgcn_wmma_f32_16x16x4_f32(
            false, aX, false, b2, (short)0, accr[ct], false, false);
      }
    }
  }

  // bias + row sum-of-squares (C layout: vgpr v -> row v + half*8, col ct*16+lm)
  float p[8];
#pragma unroll
  for (int v = 0; v < 8; ++v) p[v] = 0.f;
#pragma unroll
  for (int ct = 0; ct < 8; ++ct) {
    const float bv = bl[ct * 16 + lm];
    v8f c = acc[ct];
#pragma unroll
    for (int v = 0; v < 8; ++v) {
      float y = c[v] + bv;
      c[v] = y;
      p[v] += y * y;
    }
    acc[ct] = c;
  }
#pragma unroll
  for (int v = 0; v < 8; ++v) {   // reduce over the 16 lanes of this half-wave
    float s = p[v];
    s += __shfl_xor(s, 1, 32);
    s += __shfl_xor(s, 2, 32);
    s += __shfl_xor(s, 4, 32);
    s += __shfl_xor(s, 8, 32);
    p[v] = 1.f / fmaxf(sqrtf(s), 1e-12f);
  }

  // l2-normalize, eval-BN, residual, relu, store
#pragma unroll
  for (int ct = 0; ct < 8; ++ct) {
    const int col = ct * 16 + lm;
    const float mu = mean[col];
    const float gs = gamma[col] * rsqrtf(var[col] + 1e-5f);
    const float be = beta[col];
    const float pb = HASPROJ ? projb[col] : 0.f;
    v8f c = acc[ct];
#pragma unroll
    for (int v = 0; v < 8; ++v) {
      const int r2 = m0 + half * 8 + v;
      const int r2c = (r2 < n) ? r2 : (n - 1);
      float y = c[v] * p[v];
      y = (y - mu) * gs + be;
      float res;
      if (HASPROJ) {
        res = accr[ct][v] + pb;
      } else {
        res = xres[(size_t)r2c * 128 + col];  // clamped, unconditional load
      }
      y += res;
      if (do_relu) y = fmaxf(y, 0.f);
      if (r2 < n) out[(size_t)r2 * 128 + col] = y;
    }
  }
}

// ----------------------------------------------- cluster scoring MLP 128->64->1
__global__ void cluster_scores(const float* __restrict__ x,   // [n,128]
                               const float* __restrict__ W0,  // [128,64]
                               const float* __restrict__ b0,  // [64]
                               const float* __restrict__ W1,  // [64,1]
                               const float* __restrict__ b1,  // [1]
                               float* __restrict__ scores, int n) {
  const int node = (blockIdx.x * blockDim.x + threadIdx.x) >> 5;
  const int lane = threadIdx.x & 31;
  if (node >= n) return;
  float xr[4];
#pragma unroll
  for (int t = 0; t < 4; ++t) xr[t] = x[(size_t)node * 128 + t * 32 + lane];
  float acc = 0.f;
#pragma unroll
  for (int jj = 0; jj < 2; ++jj) {
    const int j = jj * 32 + lane;
    float h = b0[j];
#pragma unroll
    for (int t = 0; t < 4; ++t) {
      float xv = xr[t];
#pragma unroll
      for (int bb = 0; bb < 32; ++bb) {
        float xk = __shfl(xv, bb, 32);
        h += xk * W0[(t * 32 + bb) * 64 + j];
      }
    }
    h = fmaxf(h, 0.f);
    acc += h * W1[j];
  }
  for (int off = 16; off >= 1; off >>= 1) acc += __shfl_xor(acc, off, 32);
  if (lane == 0) scores[node] = acc + b1[0];
}

// -------------------------- exact top-k by rank counting (jax top_k tie rule)
__global__ void rank_topk(const float* __restrict__ scores,
                          int* __restrict__ mask, int n, int k) {
  __shared__ float sch[1024];
  const int i = blockIdx.x * 256 + threadIdx.x;
  const float si = (i < n) ? scores[i] : -3.402823e38f;
  int rank = 0;
  for (int base = 0; base < n; base += 1024) {
    for (int t = threadIdx.x; t < 1024; t += 256) {
      int j = base + t;
      sch[t] = (j < n) ? scores[j] : -3.402823e38f;
    }
    __syncthreads();
    const int lim = min(1024, n - base);
    for (int t = 0; t < lim; ++t) {
      float s = sch[t];
      int j = base + t;
      rank += ((s > si) || (s == si && j < i)) ? 1 : 0;
    }
    __syncthreads();
  }
  if (i < n) mask[i] = (rank < k) ? 1 : 0;
}

// -------------------------------------- single-block exclusive scan of mask
__global__ void scan_mask(const int* __restrict__ mask, int* __restrict__ nidx,
                          int n) {
  __shared__ int sh[1024];
  __shared__ int carrysh;
  const int tid = threadIdx.x;
  if (tid == 0) carrysh = 0;
  __syncthreads();
  for (int base = 0; base < n; base += 1024) {
    int i = base + tid;
    int v = (i < n) ? mask[i] : 0;
    sh[tid] = v;
    __syncthreads();
    for (int off = 1; off < 1024; off <<= 1) {
      int t = (tid >= off) ? sh[tid - off] : 0;
      __syncthreads();
      sh[tid] += t;
      __syncthreads();
    }
    if (i < n) nidx[i] = carrysh + sh[tid] - v;   // exclusive prefix
    __syncthreads();
    if (tid == 0) carrysh += sh[1023];
    __syncthreads();
  }
}

// --------------------------------------------------------- pooling / unpool
__global__ void pool_gather(const float* __restrict__ xin,
                            const int* __restrict__ mask,
                            const int* __restrict__ nidx,
                            float* __restrict__ pooled, int n) {
  int tid = blockIdx.x * blockDim.x + threadIdx.x;
  int i = tid >> 5, q = tid & 31;
  if (i >= n || !mask[i]) return;
  const float4* s4 = reinterpret_cast<const float4*>(xin);
  float4* d4 = reinterpret_cast<float4*>(pooled);
  d4[(size_t)nidx[i] * 32 + q] = s4[(size_t)i * 32 + q];
}

__global__ void unpool_concat(const float* __restrict__ localf,
                              const float* __restrict__ globf,
                              const int* __restrict__ mask,
                              const int* __restrict__ nidx,
                              float* __restrict__ comb, int n) {
  int tid = blockIdx.x * blockDim.x + threadIdx.x;
  int i = tid >> 6, q = tid & 63;
  if (i >= n) return;
  const float4* l4 = reinterpret_cast<const float4*>(localf);
  const float4* g4 = reinterpret_cast<const float4*>(globf);
  float4* c4 = reinterpret_cast<float4*>(comb);
  if (q < 32) {
    c4[(size_t)i * 64 + q] = l4[(size_t)i * 32 + q];
  } else {
    int gi = mask[i] ? nidx[i] : 0;   // dropped nodes map to cluster 0
    c4[(size_t)i * 64 + q] = g4[(size_t)gi * 32 + (q - 32)];
  }
}

// ---------------------------------------------------------------- launcher
extern "C" void kernel_launch(void* const* d_in, const int* in_sizes, int n_in,
                              void* d_out, int out_size, void* d_ws,
                              size_t ws_size, hipStream_t stream) {
  (void)n_in; (void)out_size; (void)ws_size;
  const float* x0 = (const float*)d_in[0];
  const int* ei = (const int*)d_in[1];
  const int N = in_sizes[0] / 128;
  const int E = in_sizes[1] / 2;
  const int K = (N / 2) > 1 ? (N / 2) : 1;
  const int* src = ei;
  const int* dst = ei + E;

  auto P = [&](int i) { return (const float*)d_in[i]; };

  // workspace layout (floats)
  float* ws = (float*)d_ws;
  size_t off = 0;
  float* agg = ws + off;    off += (size_t)N * 256;
  float* cnt = ws + off;    off += (size_t)N;
  float* xa = ws + off;     off += (size_t)N * 256;
  float* xb = ws + off;     off += (size_t)N * 256;
  float* localK = ws + off; off += (size_t)N * 128;
  float* globK = ws + off;  off += (size_t)N * 128;
  float* scores = ws + off; off += (size_t)N;
  int* maskb = (int*)(ws + off); off += (size_t)N;
  int* nidx = (int*)(ws + off);  off += (size_t)N;

  const dim3 B(256);
  auto blocks = [](size_t t) { return dim3((unsigned)((t + 255) / 256)); };

  auto run_layer = [&](const float* xin, int din, int nrows, const int* mask,
                       const int* remap, int pbase, bool hasproj, int pw,
                       int pb2, float* outp, int relu) {
    fill_f32<<<blocks((size_t)nrows * din), B, 0, stream>>>(agg, 0.f,
                                                            (size_t)nrows * din);
    fill_f32<<<blocks((size_t)nrows), B, 0, stream>>>(cnt, 0.f, (size_t)nrows);
    const int fshift = (din == 256) ? 6 : 5;
    scatter_mean_acc<<<blocks((size_t)E << fshift), B, 0, stream>>>(
        xin, src, dst, mask, remap, E, fshift, agg, cnt);
    const size_t waves = ((size_t)nrows + 15) / 16;
    if (hasproj) {
      sage_layer_gemm<true><<<blocks(waves * 32), B, 0, stream>>>(
          agg, cnt, xin, P(pbase + 0), P(pbase + 1), P(pbase + 2),
          P(pbase + 3), P(pbase + 4), P(pbase + 5), P(pbase + 6), xin, P(pw),
          P(pb2), outp, nrows, din, relu);
    } else {
      sage_layer_gemm<false><<<blocks(waves * 32), B, 0, stream>>>(
          agg, cnt, xin, P(pbase + 0), P(pbase + 1), P(pbase + 2),
          P(pbase + 3), P(pbase + 4), P(pbase + 5), P(pbase + 6), xin, nullptr,
          nullptr, outp, nrows, din, relu);
    }
  };

  // 1) local encoder (3 layers, full graph), params at 2, 9, 16
  run_layer(x0, 128, N, nullptr, nullptr, 2, false, 0, 0, xa, 1);
  run_layer(xa, 128, N, nullptr, nullptr, 9, false, 0, 0, xb, 1);
  run_layer(xb, 128, N, nullptr, nullptr, 16, false, 0, 0, localK, 0);

  // 2) top-k pooling (cluster params at 53..56)
  cluster_scores<<<blocks((size_t)N * 32), B, 0, stream>>>(
      localK, P(53), P(54), P(55), P(56), scores, N);
  rank_topk<<<blocks((size_t)N), B, 0, stream>>>(scores, maskb, N, K);
  scan_mask<<<1, 1024, 0, stream>>>(maskb, nidx, N);
  pool_gather<<<blocks((size_t)N * 32), B, 0, stream>>>(localK, maskb, nidx,
                                                        xa, N);

  // 3) global encoder on pooled graph (params at 23, 30)
  run_layer(xa, 128, K, maskb, nidx, 23, false, 0, 0, xb, 1);
  run_layer(xb, 128, K, maskb, nidx, 30, false, 0, 0, globK, 0);

  // 4) unpool + concat -> [local | glob[new_idx]]
  unpool_concat<<<blocks((size_t)N * 64), B, 0, stream>>>(localK, globK, maskb,
                                                          nidx, xa, N);

  // 5) refine encoder (layer0 din=256 with proj at 44/45; layer1 at 46)
  run_layer(xa, 256, N, nullptr, nullptr, 37, true, 44, 45, xb, 1);
  run_layer(xb, 128, N, nullptr, nullptr, 46, false, 0, 0, (float*)d_out, 0);
}